// ErosionLayer_59304908423804
// MI455X (gfx1250) — compile-verified
//
#include <hip/hip_runtime.h>
#include <cstdint>
#include <cstddef>

#define W       512
#define NITERS  10
#define NB      16
#define TILE    32
#define LROWS   36                 // rows rBase-2 .. rBase+33
#define LCOLS   40                 // cols cBase-4 .. cBase+35 (16B-aligned start)
#define CELLW   0.390625f          // 200/512, exact in fp32

typedef __attribute__((ext_vector_type(4))) unsigned u32x4;
typedef __attribute__((ext_vector_type(8))) unsigned u32x8;

__device__ __forceinline__ float relu(float x) { return fmaxf(x, 0.0f); }

// --- CDNA5 async global -> LDS element copy (ASYNCcnt path) -----------------
// Generic pointers to __shared__ carry the LDS byte offset in their low 32 bits
// (aperture is in bits 63:32 per ISA 10.2), which is what VDST expects.
__device__ __forceinline__ void async_to_lds_b32(void* ldsPtr, const float* g) {
    unsigned l = (unsigned)(unsigned long long)ldsPtr;
    asm volatile("global_load_async_to_lds_b32 %0, %1, off"
                 :: "v"(l), "v"(g) : "memory");
}
__device__ __forceinline__ void wait_async0() {
#if __has_builtin(__builtin_amdgcn_s_wait_asynccnt)
    __builtin_amdgcn_s_wait_asynccnt(0);
#else
    asm volatile("s_wait_asynccnt 0" ::: "memory");
#endif
}

// --- CDNA5 Tensor Data Mover: DMA one 36x40-f32 tile (row stride 512) -------
// D# per CDNA5 ISA; 2D tile, data_size=4B, no padding/iterate,
// workgroup_mask=0 (not in a cluster). tensor_dim2/3/4 = 1 so no dim flags OOB.
__device__ __forceinline__ void tdm_load_tile(void* ldsPtr, const float* gsrc) {
    const unsigned long long ga = (unsigned long long)gsrc;
    u32x4 g0;
    g0[0] = 1u;                                            // count=1, user desc
    g0[1] = (unsigned)(unsigned long long)ldsPtr;          // lds_addr (bytes)
    g0[2] = (unsigned)ga;                                  // global_addr[31:0]
    g0[3] = ((unsigned)(ga >> 32) & 0x01FFFFFFu) | (2u << 30); // addr[56:32], type=2
    u32x8 g1;
    g1[0] = 2u << 16;                  // data_size=2 (4B); mask/flags/pad = 0
    g1[1] = (unsigned)LCOLS << 16;     // tensor_dim0 = 40  (bits 79:48)
    g1[2] = (unsigned)LROWS << 16;     // tensor_dim1 = 36  (bits 111:80)
    g1[3] = (unsigned)LCOLS << 16;     // tile_dim0   = 40  (bits 127:112)
    g1[4] = (unsigned)LROWS;           // tile_dim1 = 36, tile_dim2 = 0
    g1[5] = (unsigned)W;               // tensor_dim0_stride = 512 (elements)
    g1[6] = 0u;                        // stride0 hi / tensor_dim1_stride lo
    g1[7] = 0u;
    u32x4 g2; g2[0] = 1u; g2[1] = 1u; g2[2] = 0u; g2[3] = 0u;   // dim2=1, dim3=1
    u32x4 g3; g3[0] = 0u; g3[1] = 1u << 16; g3[2] = 0u; g3[3] = 0u; // dim4=1
    asm volatile("tensor_load_to_lds %0, %1, %2, %3"
                 :: "s"(g0), "s"(g1), "s"(g2), "s"(g3)
                 : "memory");
}

// -------------------- init: terrain = (1-x)/2, states = 0 -------------------
__global__ __launch_bounds__(256) void erosion_init(const float4* __restrict__ in,
                                                    float4* __restrict__ terr,
                                                    float4* __restrict__ sed,
                                                    float4* __restrict__ wat,
                                                    float4* __restrict__ vel, int n4) {
    int i = blockIdx.x * 256 + threadIdx.x;
    if (i < n4) {
        float4 x = in[i];
        float4 t = make_float4((1.0f - x.x) * 0.5f, (1.0f - x.y) * 0.5f,
                               (1.0f - x.z) * 0.5f, (1.0f - x.w) * 0.5f);
        float4 z = make_float4(0.f, 0.f, 0.f, 0.f);
        terr[i] = t; sed[i] = z; wat[i] = z; vel[i] = z;
    }
}

// -------------------- one erosion step --------------------
__global__ __launch_bounds__(256) void erosion_step(
    const float* __restrict__ tin_all, float* __restrict__ tout_all,
    float* __restrict__ sed_all, float* __restrict__ wat_all, float* __restrict__ vel_all,
    const float* __restrict__ rain, const float* __restrict__ gnoise,
    const float* __restrict__ pRain, const float* __restrict__ pEvap,
    const float* __restrict__ pMhd,  const float* __restrict__ pHeps,
    const float* __restrict__ pGrav, const float* __restrict__ pScc,
    const float* __restrict__ pDiss, const float* __restrict__ pDepo)
{
    __shared__ __align__(16) float tile[LROWS * LCOLS];

    const int tid   = threadIdx.x;
    const int cBase = blockIdx.x * TILE;           // j (axis 2)
    const int rBase = blockIdx.y * TILE;           // i (axis 1)
    const size_t base = (size_t)blockIdx.z * W * W;
    const float* tin = tin_all + base;

    // gfx1250 prefetch of the streamed per-cell state for this tile
    __builtin_prefetch(sed_all + base + (size_t)rBase * W + cBase, 0, 3);
    __builtin_prefetch(wat_all + base + (size_t)rBase * W + cBase, 0, 3);
    __builtin_prefetch(vel_all + base + (size_t)rBase * W + cBase, 0, 3);

    const bool interior = (blockIdx.x >= 1) && (blockIdx.x <= (W / TILE) - 2) &&
                          (blockIdx.y >= 1) && (blockIdx.y <= (W / TILE) - 2);
    if (interior) {
        // Whole 36x40 halo tile is in-bounds: one TDM DMA per block, wave 0 only
        // (TDM ignores EXEC; per-wave issue, per-wave TENSORcnt).
        if (tid < 32) {
            tdm_load_tile(tile, tin + (size_t)(rBase - 2) * W + (cBase - 4));
            __builtin_amdgcn_s_wait_tensorcnt(0);
        }
    } else {
        // Border blocks: per-element clamped async copies (clamped halo values
        // are never consumed: boundary formulas + bilinear valid-mask).
        for (int e = tid; e < LROWS * LCOLS; e += 256) {
            int lr = e / LCOLS;
            int lc = e - lr * LCOLS;
            int gr = min(max(rBase - 2 + lr, 0), W - 1);
            int gc = min(max(cBase - 4 + lc, 0), W - 1);
            async_to_lds_b32(&tile[lr * LCOLS + lc], tin + gr * W + gc);
        }
        wait_async0();
    }
    __syncthreads();

    const float rainRate = relu(pRain[0]);
    const float evap     = relu(pEvap[0]);
    const float mhd      = pMhd[0];
    const float heps     = pHeps[0];
    const float grav     = relu(pGrav[0]);
    const float scc      = relu(pScc[0]);
    const float diss     = pDiss[0];
    const float depo     = pDepo[0];

    const int lj = tid & (TILE - 1);
    const int ty = tid >> 5;          // wave id 0..7 (wave32)
    const int j  = cBase + lj;

#define AT(gi, gj) tile[((gi) - rBase + 2) * LCOLS + ((gj) - cBase + 4)]

#pragma unroll
    for (int k = 0; k < 4; ++k) {
        const int i = rBase + ty + 8 * k;
        const float t = AT(i, j);

        // simple_gradient (dx along axis1 = i, dy along axis2 = j)
        float dx, dy;
        if (i == 0)          dx = 0.05f * t;
        else if (i == W - 1) dx = -0.05f * t;
        else                 dx = 0.5f * (AT(i + 1, j) - AT(i - 1, j));
        if (j == 0)          dy = 0.05f * t;
        else if (j == W - 1) dy = -0.05f * t;
        else                 dy = 0.5f * (AT(i, j + 1) - AT(i, j - 1));

        const float mag    = sqrtf(dx * dx + dy * dy + 1e-11f);
        const float rX     = gnoise[i * W + j];
        const float rY     = sqrtf(1.0f - rX * rX);
        const float factor = relu(1e-10f - mag);
        const float inv    = 1.0f / (mag + factor);
        const float fdx    = (dx + factor * rX) * inv;
        const float fdy    = (dy + factor * rY) * inv;

        // bilinear_sample(terrain, -gradient): x offset on col j, y offset on row i
        const float fx  = (float)j - fdx;
        const float fy  = (float)i - fdy;
        const float x0f = floorf(fx), y0f = floorf(fy);
        const float wx1 = fx - x0f,   wy1 = fy - y0f;
        const int   x0  = (int)x0f,   y0  = (int)y0f;

        auto g = [&](int ix, int iy) -> float {
            if (ix < 0 || ix > W - 1 || iy < 0 || iy > W - 1) return 0.0f;
            return AT(iy, ix) - 1.0f;   // tm1 = t - 1
        };
        const float nb =
            (1.0f - wy1) * ((1.0f - wx1) * g(x0, y0)     + wx1 * g(x0 + 1, y0)) +
                    wy1  * ((1.0f - wx1) * g(x0, y0 + 1) + wx1 * g(x0 + 1, y0 + 1)) + 1.0f;

        const float hd  = t - nb;
        const float hds = (hd - heps > 0.0f) ? 1.0f : 0.0f;      // sign(relu(hd-eps))
        const float nhd = hds * fmaxf(hd, mhd);

        const size_t idx = base + (size_t)i * W + j;
        float wtr = wat_all[idx] + rainRate * rain[i * W + j];
        const float v0  = vel_all[idx];
        float sed = sed_all[idx];

        const float sed_cap = nhd / CELLW * v0 * wtr * scc;
        const float ftb   = (hd < 0.0f) ? 1.0f : 0.0f;           // relu(sign(-hd))
        const float first = fminf(relu(-hd), sed);
        const float sdiff = sed - sed_cap;
        const float third = (1.0f - ftb) *
                            (relu(sdiff * depo) - relu(-sdiff * diss));
        const float dep   = fmaxf(-relu(hd), first + third);

        sed -= dep;
        const float tNew = t + dep;

        // displace (per-cell as written; only boundary columns masked)
        float disp = relu(1.0f - fabsf(fdy));
        if (j != W - 1) disp += relu(-fdy);
        if (j != 0)     disp += relu(fdy);
        sed *= disp;
        wtr *= disp;

        const float vNew = grav * hd / CELLW;
        wtr *= (1.0f - evap);

        tout_all[idx] = tNew;
        sed_all[idx]  = sed;
        wat_all[idx]  = wtr;
        vel_all[idx]  = vNew;
    }
#undef AT
}

// -------------------- finalize in place: relu(2 - 2t) - 1 -------------------
__global__ __launch_bounds__(256) void erosion_final(float4* __restrict__ t, int n4) {
    int i = blockIdx.x * 256 + threadIdx.x;
    if (i < n4) {
        float4 v = t[i];
        v.x = fmaxf(2.0f - 2.0f * v.x, 0.0f) - 1.0f;
        v.y = fmaxf(2.0f - 2.0f * v.y, 0.0f) - 1.0f;
        v.z = fmaxf(2.0f - 2.0f * v.z, 0.0f) - 1.0f;
        v.w = fmaxf(2.0f - 2.0f * v.w, 0.0f) - 1.0f;
        t[i] = v;
    }
}

extern "C" void kernel_launch(void* const* d_in, const int* in_sizes, int n_in,
                              void* d_out, int out_size, void* d_ws, size_t ws_size,
                              hipStream_t stream) {
    const float* inp   = (const float*)d_in[0];
    const float* rain  = (const float*)d_in[1];
    const float* gnz   = (const float*)d_in[2];
    const float* pRain = (const float*)d_in[3];
    const float* pEvap = (const float*)d_in[4];
    const float* pMhd  = (const float*)d_in[5];
    const float* pHeps = (const float*)d_in[6];
    const float* pGrav = (const float*)d_in[7];
    const float* pScc  = (const float*)d_in[8];
    const float* pDiss = (const float*)d_in[9];
    const float* pDepo = (const float*)d_in[10];

    const size_t N = (size_t)NB * W * W;           // 4,194,304 cells
    float* tA  = (float*)d_out;                    // ping buffer 0 == output
    float* ws  = (float*)d_ws;
    float* tB  = ws;                               // ping buffer 1
    float* sed = ws + N;
    float* wat = ws + 2 * N;
    float* vel = ws + 3 * N;

    const int n4 = (int)(N / 4);
    const dim3 eb(256);
    const dim3 eg((unsigned)((n4 + 255) / 256));
    erosion_init<<<eg, eb, 0, stream>>>((const float4*)inp, (float4*)tA,
                                        (float4*)sed, (float4*)wat, (float4*)vel, n4);

    const dim3 grid(W / TILE, W / TILE, NB), block(256);
    for (int it = 0; it < NITERS; ++it) {
        const float* tin = (it & 1) ? tB : tA;
        float*       tout = (it & 1) ? tA : tB;
        erosion_step<<<grid, block, 0, stream>>>(
            tin, tout, sed, wat, vel,
            rain + (size_t)it * W * W, gnz + (size_t)it * W * W,
            pRain, pEvap, pMhd, pHeps, pGrav, pScc, pDiss, pDepo);
    }
    // after 10 steps (even count) result sits in tA == d_out
    erosion_final<<<eg, eb, 0, stream>>>((float4*)tA, n4);
}